// TriplaneFeatureNet_6511170421531
// MI455X (gfx1250) — compile-verified
//
#include <hip/hip_runtime.h>
#include <cstdint>

// ---------------------------------------------------------------------------
// TriplaneFeatureNet for MI455X (gfx1250)
//
// Roofline: output = 540 MB streamed (~25us @ 23.3 TB/s HBM) is the floor.
// Triplane (100 MB) fits in the 192 MB L2 -> keep it hot there; write all
// output with nontemporal (TH_STORE_NT) stores so the 540 MB stream doesn't
// evict it. Native [C,512,512] layout puts a texel's 32 channels 1 MB apart
// (32 lines per corner fetch) -> transpose once to channels-last
// [3,512,512,32] in d_ws so each bilinear corner is ONE coalesced 128 B load
// per wave (lane == channel, wave32).
//
// CDNA5 paths: global_load_async_to_lds_b32 + s_wait_asynccnt (async DMA into
// LDS) in the transpose; NT stores in the sampler; readfirstlane-forced
// scalarization so corner loads/stores use SGPR-base (saddr) addressing and
// point loads become scalar (constant-cache) loads.
// ---------------------------------------------------------------------------

#define RES 512
#define FEAT 32
#define OUT_STRIDE 135   // 32*3 + 39
#define N_FREQS 6

// ---------------- transpose: [3][32][512][512] -> [3][512][512][32] --------
// grid: 3 * 512 * 16 blocks, 256 threads. Each block: one (plane,row,32-x tile)
// = 32 channels x 32 x positions = 4 KB, loaded via async DMA into LDS.
__global__ __launch_bounds__(256) void triplane_transpose_kernel(
    const float* __restrict__ in, float* __restrict__ out) {
  __shared__ float tile[32][33];  // +1 pad: column/row accesses conflict-free

  const int b  = blockIdx.x;
  const int xt = b & 15;           // x tile (16 tiles of 32)
  const int y  = (b >> 4) & (RES - 1);
  const int pl = b >> 13;          // plane
  const int x0 = xt * 32;
  const int t  = threadIdx.x;

  // ---- async load phase: 1024 floats, 4 per thread, row-coalesced ----
#pragma unroll
  for (int it = 0; it < 4; ++it) {
    const int i  = it * 256 + t;
    const int c  = i >> 5;   // channel (row of tile); uniform within a wave
    const int xi = i & 31;   // lane-contiguous -> 128B coalesced per wave
    const float* gsrc =
        in + (((size_t)(pl * FEAT + c) * RES + y) * RES + (x0 + xi));
    const unsigned lds_addr = (unsigned)(uintptr_t)(&tile[c][xi]);
    asm volatile("global_load_async_to_lds_b32 %0, %1, off"
                 :
                 : "v"(lds_addr), "v"(gsrc)
                 : "memory");
  }
  asm volatile("s_wait_asynccnt 0" ::: "memory");
  __syncthreads();

  // ---- store phase: lane = channel -> contiguous 128B per wave ----
  const int c2  = t & 31;
  const int xig = t >> 5;  // 0..7
  float* obase = out + ((size_t)(pl * RES + y) * RES) * FEAT;
#pragma unroll
  for (int q = 0; q < 4; ++q) {
    const int xi = xig * 4 + q;
    obase[(size_t)(x0 + xi) * FEAT + c2] = tile[c2][xi];  // RT: keep in L2
  }
}

// ---------------- sampler: one point per wave32, lane = channel ------------
// All per-point math is wave-uniform; readfirstlane pushes it into SGPRs so
// gathers/stores use saddr+lane addressing and point loads become s_loads.
template <bool TRANSPOSED>
__global__ __launch_bounds__(256) void triplane_sample_kernel(
    const float* __restrict__ pts, const float* __restrict__ tp,
    const float* __restrict__ aabb, float* __restrict__ out, int n_points) {
  const int wv   = __builtin_amdgcn_readfirstlane(threadIdx.x >> 5);  // SGPR
  const int lane = threadIdx.x & 31;
  const int wave = blockIdx.x * 8 + wv;  // scalar point index
  if (wave >= n_points) return;

  // point coords: scalar address -> constant-cache loads, broadcast to lanes
  float xyz[3];
#pragma unroll
  for (int d = 0; d < 3; ++d) {
    const float a0 = aabb[d], a1 = aabb[3 + d];
    float tc = (pts[(size_t)wave * 3 + d] - a0) / (a1 - a0);
    tc = fminf(fmaxf(tc, 0.0f), 1.0f);
    xyz[d] = 2.0f * tc - 1.0f;
  }

  float* optr = out + (size_t)wave * OUT_STRIDE;  // scalar output base

  // plane -> (gx, gy) dims: xy:(x,y)  yz:(y,z)  xz:(x,z)
  const int gxd[3] = {0, 1, 0};
  const int gyd[3] = {1, 2, 2};

#pragma unroll
  for (int pl = 0; pl < 3; ++pl) {
    const float gx = xyz[gxd[pl]];
    const float gy = xyz[gyd[pl]];
    const float fx = (gx + 1.0f) * 0.5f * (float)(RES - 1);
    const float fy = (gy + 1.0f) * 0.5f * (float)(RES - 1);
    const float x0f = floorf(fx), y0f = floorf(fy);
    const float wx = fx - x0f, wy = fy - y0f;
    const int x0 = min(max((int)x0f, 0), RES - 1);
    const int x1 = min(x0 + 1, RES - 1);
    const int y0 = min(max((int)y0f, 0), RES - 1);
    const int y1 = min(y0 + 1, RES - 1);

    float v00, v01, v10, v11;
    if (TRANSPOSED) {
      // channels-last: scalar corner base + lane*4 -> one 128B txn per corner
      const float* pb = tp + ((size_t)pl * RES * RES) * FEAT;
      const int r0 = y0 * (RES * FEAT), r1 = y1 * (RES * FEAT);
      const int cx0 = x0 * FEAT, cx1 = x1 * FEAT;
      const unsigned o00 = __builtin_amdgcn_readfirstlane(r0 + cx0);
      const unsigned o01 = __builtin_amdgcn_readfirstlane(r0 + cx1);
      const unsigned o10 = __builtin_amdgcn_readfirstlane(r1 + cx0);
      const unsigned o11 = __builtin_amdgcn_readfirstlane(r1 + cx1);
      v00 = (pb + o00)[lane];
      v01 = (pb + o01)[lane];
      v10 = (pb + o10)[lane];
      v11 = (pb + o11)[lane];
    } else {
      // fallback: native [C,R,R]; scalar (y,x) base + lane*R*R element offset
      const float* pb = tp + (size_t)pl * FEAT * RES * RES;
      const unsigned b00 = __builtin_amdgcn_readfirstlane(y0 * RES + x0);
      const unsigned b01 = __builtin_amdgcn_readfirstlane(y0 * RES + x1);
      const unsigned b10 = __builtin_amdgcn_readfirstlane(y1 * RES + x0);
      const unsigned b11 = __builtin_amdgcn_readfirstlane(y1 * RES + x1);
      const int lo = lane * (RES * RES);
      v00 = (pb + b00)[lo];
      v01 = (pb + b01)[lo];
      v10 = (pb + b10)[lo];
      v11 = (pb + b11)[lo];
    }

    const float top = v00 * (1.0f - wx) + v01 * wx;
    const float bot = v10 * (1.0f - wx) + v11 * wx;
    const float f   = top * (1.0f - wy) + bot * wy;
    // streamed output: NT hint so 540 MB of writes don't evict L2 triplane
    __builtin_nontemporal_store(f, optr + pl * FEAT + lane);
  }

  // ---- positional encoding: 39 values; reshape order [freq][sin,cos][dim] --
  auto posenc_val = [&](int k) -> float {
    if (k < 3) return xyz[k];
    const int j = k - 3;
    const int f = j / 6;
    const int rem = j % 6;
    const int s = rem / 3;   // 0 = sin, 1 = cos
    const int d = rem % 3;
    const float a = xyz[d] * (float)(1 << f);
    const float sv = sinf(a);
    const float cv = cosf(a);
    return s ? cv : sv;
  };

  __builtin_nontemporal_store(posenc_val(lane), optr + 96 + lane);
  if (lane < OUT_STRIDE - 96 - 32) {  // lanes 0..6 cover k = 32..38
    __builtin_nontemporal_store(posenc_val(lane + 32), optr + 96 + 32 + lane);
  }
}

extern "C" void kernel_launch(void* const* d_in, const int* in_sizes, int n_in,
                              void* d_out, int out_size, void* d_ws,
                              size_t ws_size, hipStream_t stream) {
  const float* p    = (const float*)d_in[0];
  const float* tp   = (const float*)d_in[1];
  const float* aabb = (const float*)d_in[2];
  float* out = (float*)d_out;
  const int n = in_sizes[0] / 3;

  const size_t WS_BYTES = (size_t)3 * RES * RES * FEAT * sizeof(float);  // 100 MB
  const int sample_blocks = (n + 7) / 8;  // 8 waves (points) per 256-thr block

  if (ws_size >= WS_BYTES) {
    float* wsf = (float*)d_ws;
    triplane_transpose_kernel<<<3 * RES * 16, 256, 0, stream>>>(tp, wsf);
    triplane_sample_kernel<true>
        <<<sample_blocks, 256, 0, stream>>>(p, wsf, aabb, out, n);
  } else {
    triplane_sample_kernel<false>
        <<<sample_blocks, 256, 0, stream>>>(p, tp, aabb, out, n);
  }
}